// AggregationLayer_53051436040325
// MI455X (gfx1250) — compile-verified
//
#include <hip/hip_runtime.h>

#define DFEAT 32           // feature dim (fixed by reference)
#define BLOCK 256          // 8 wave32 waves per block

// One wave per output segment. Lane l owns feature column l (D == warpSize == 32),
// so each gathered src row is exactly one 128B coalesced cacheline transaction.
// segment_ids are sorted, so segment n occupies a contiguous edge range found by
// binary search -> no atomics, fully deterministic, single pass.
// src (12.8MB) is L2-resident on MI455X (192MB L2): the gather stream is L2
// bandwidth, so we maximize in-flight cacheline gathers per wave (8-wide) and
// keep the output store non-temporal to avoid polluting L2.
__global__ __launch_bounds__(BLOCK) void AggregationLayer_segmean_kernel(
    const float* __restrict__ src,        // [N, 32] f32
    const int*   __restrict__ gidx,       // [E] gather indices
    const int*   __restrict__ segid,      // [E] sorted segment ids
    float*       __restrict__ out,        // [N, 32] f32
    int N, int E)
{
    const int lane = threadIdx.x & 31;
    int wave = (int)((blockIdx.x * (unsigned)blockDim.x + threadIdx.x) >> 5);
    // Force the segment id into an SGPR: binary search + index loads become
    // scalar (SMEM) loads instead of 32 redundant VMEM lanes.
    const int n = __builtin_amdgcn_readfirstlane(wave);
    if (n >= N) return;

    // lower_bound(segid, n)
    int lo = 0, hi = E;
    while (lo < hi) {
        int mid = (lo + hi) >> 1;
        if (segid[mid] < n) lo = mid + 1; else hi = mid;
    }
    const int start = lo;
    // lower_bound(segid, n+1)  (lo already >= start)
    hi = E;
    while (lo < hi) {
        int mid = (lo + hi) >> 1;
        if (segid[mid] < n + 1) lo = mid + 1; else hi = mid;
    }
    const int end = lo;

    // Stream the segment's edges: per edge, the 32 lanes fetch one 128B row.
    // 8-way unroll keeps 8 independent cacheline gathers in flight per wave;
    // global_prefetch_b8 (gfx1250) warms rows 16 edges ahead of the front.
    float s0 = 0.f, s1 = 0.f, s2 = 0.f, s3 = 0.f;
    float s4 = 0.f, s5 = 0.f, s6 = 0.f, s7 = 0.f;
    int e = start;
    for (; e + 8 <= end; e += 8) {
        const int i0 = gidx[e + 0];
        const int i1 = gidx[e + 1];
        const int i2 = gidx[e + 2];
        const int i3 = gidx[e + 3];
        const int i4 = gidx[e + 4];
        const int i5 = gidx[e + 5];
        const int i6 = gidx[e + 6];
        const int i7 = gidx[e + 7];
        const int ep = e + 16;
        if (ep < end) {
            const int ip = gidx[ep];
            __builtin_prefetch(&src[(size_t)ip * DFEAT + lane], 0, 3);
        }
        s0 += src[(size_t)i0 * DFEAT + lane];
        s1 += src[(size_t)i1 * DFEAT + lane];
        s2 += src[(size_t)i2 * DFEAT + lane];
        s3 += src[(size_t)i3 * DFEAT + lane];
        s4 += src[(size_t)i4 * DFEAT + lane];
        s5 += src[(size_t)i5 * DFEAT + lane];
        s6 += src[(size_t)i6 * DFEAT + lane];
        s7 += src[(size_t)i7 * DFEAT + lane];
    }
    for (; e < end; ++e)
        s0 += src[(size_t)gidx[e] * DFEAT + lane];

    const float sum = ((s0 + s1) + (s2 + s3)) + ((s4 + s5) + (s6 + s7));
    const float cnt = (float)(end - start);
    // Empty segments: sum==0, max(cnt,1)==1 -> 0, matching the reference.
    // Non-temporal store: output is write-once, keep L2 free for src rows.
    __builtin_nontemporal_store(sum / fmaxf(cnt, 1.0f),
                                &out[(size_t)n * DFEAT + lane]);
}

extern "C" void kernel_launch(void* const* d_in, const int* in_sizes, int n_in,
                              void* d_out, int out_size, void* d_ws, size_t ws_size,
                              hipStream_t stream) {
    (void)n_in; (void)out_size; (void)d_ws; (void)ws_size;
    const float* src   = (const float*)d_in[0];   // [N*32] f32
    const int*   gidx  = (const int*)d_in[1];     // [E] i32
    const int*   segid = (const int*)d_in[2];     // [E] i32 (sorted)
    float*       out   = (float*)d_out;           // [N*32] f32

    const int N = in_sizes[0] / DFEAT;
    const int E = in_sizes[1];

    const int wavesPerBlock = BLOCK / 32;
    const int nBlocks = (N + wavesPerBlock - 1) / wavesPerBlock;
    AggregationLayer_segmean_kernel<<<nBlocks, BLOCK, 0, stream>>>(
        src, gidx, segid, out, N, E);
}